// AttentionBlock_1795296329997
// MI455X (gfx1250) — compile-verified
//
#include <hip/hip_runtime.h>
#include <math.h>

typedef __attribute__((ext_vector_type(2))) float v2f;
typedef __attribute__((ext_vector_type(8))) float v8f;

#define NUM_HEADS 16
#define HEAD_DIM  64
#define SEQ       2048
#define BATCH     2
#define DMODEL    1024
#define NPROJ     3072  // NUM_HEADS * 3 * HEAD_DIM

// ---------------------------------------------------------------------------
// Native fp32 WMMA: D(16x16) = A(16x4) * B(4x16) + C   (wave32, CDNA5)
// A frag: lane l<16 -> row l, K = {0,1}; lane l>=16 -> row l-16, K = {2,3}
// C/D   : VGPR r, lane l -> (row r + 8*(l>=16), col l&15)
// ---------------------------------------------------------------------------
__device__ __forceinline__ v8f wmma_f32(v2f a, v2f b, v8f c) {
  return __builtin_amdgcn_wmma_f32_16x16x4_f32(false, a, false, b, (short)0, c,
                                               false, false);
}

__device__ __forceinline__ v8f v8zero() {
  v8f z;
#pragma unroll
  for (int i = 0; i < 8; ++i) z[i] = 0.0f;
  return z;
}

// ---------------------------------------------------------------------------
// Kernel 1: in_proj GEMM  C[4096,3072] = inputs[4096,1024] * w_in[1024,3072]
// Epilogue scatters columns h*192+f into Q/K/V stored [b,h,s,64] (un-roped).
// ---------------------------------------------------------------------------
__global__ __launch_bounds__(256) void gemm_in_proj(
    const float* __restrict__ A, const float* __restrict__ Bw,
    float* __restrict__ Qd, float* __restrict__ Kd, float* __restrict__ Vd) {
  __shared__ float As[64][33];
  __shared__ float Bs[32][129];
  const int tid = threadIdx.x;
  const int lane = tid & 31;
  const int wave = tid >> 5;
  const int hv = lane >> 4;
  const int ln = lane & 15;
  const int m0 = blockIdx.y * 64;
  const int n0 = blockIdx.x * 128;
  const int mw = (wave >> 2) * 32;
  const int nw = (wave & 3) * 32;

  v8f C[2][2];
  C[0][0] = v8zero(); C[0][1] = v8zero();
  C[1][0] = v8zero(); C[1][1] = v8zero();

  for (int kb = 0; kb < DMODEL; kb += 32) {
    for (int i = tid; i < 512; i += 256) {
      const int r = i >> 3, c = (i & 7) << 2;
      const float4 v = *(const float4*)(A + (size_t)(m0 + r) * DMODEL + kb + c);
      As[r][c] = v.x; As[r][c + 1] = v.y; As[r][c + 2] = v.z; As[r][c + 3] = v.w;
    }
    for (int i = tid; i < 1024; i += 256) {
      const int r = i >> 5, c = (i & 31) << 2;
      const float4 v = *(const float4*)(Bw + (size_t)(kb + r) * NPROJ + n0 + c);
      Bs[r][c] = v.x; Bs[r][c + 1] = v.y; Bs[r][c + 2] = v.z; Bs[r][c + 3] = v.w;
    }
    if (kb + 32 < DMODEL) {
      __builtin_prefetch(A + (size_t)(m0 + (tid >> 2)) * DMODEL + kb + 32, 0, 0);
      __builtin_prefetch(Bw + (size_t)(kb + 32 + (tid >> 3)) * NPROJ + n0, 0, 0);
    }
    __syncthreads();
#pragma unroll
    for (int kk = 0; kk < 32; kk += 4) {
      const int ka = kk + 2 * hv;
      v2f a0, a1, b0, b1;
      a0.x = As[mw + ln][ka];      a0.y = As[mw + ln][ka + 1];
      a1.x = As[mw + 16 + ln][ka]; a1.y = As[mw + 16 + ln][ka + 1];
      b0.x = Bs[ka][nw + ln];      b0.y = Bs[ka + 1][nw + ln];
      b1.x = Bs[ka][nw + 16 + ln]; b1.y = Bs[ka + 1][nw + 16 + ln];
      C[0][0] = wmma_f32(a0, b0, C[0][0]);
      C[0][1] = wmma_f32(a0, b1, C[0][1]);
      C[1][0] = wmma_f32(a1, b0, C[1][0]);
      C[1][1] = wmma_f32(a1, b1, C[1][1]);
    }
    __syncthreads();
  }

#pragma unroll
  for (int mi = 0; mi < 2; ++mi)
#pragma unroll
    for (int ni = 0; ni < 2; ++ni)
#pragma unroll
      for (int r = 0; r < 8; ++r) {
        const int m = m0 + mw + mi * 16 + r + 8 * hv;
        const int c = n0 + nw + ni * 16 + ln;
        const int b = m >> 11;  // /2048
        const int s = m & (SEQ - 1);
        const int h = c / 192;
        const int f = c - h * 192;
        const size_t base = ((size_t)(b * NUM_HEADS + h) * SEQ + s) * HEAD_DIM;
        const float val = C[mi][ni][r];
        if (f < 64)       Qd[base + f] = val;
        else if (f < 128) Kd[base + f - 64] = val;
        else              Vd[base + f - 128] = val;
      }
}

// ---------------------------------------------------------------------------
// Kernel 2: in-place RoPE on Q (with 1/sqrt(64) scale fused) and K.
// pair (p, p+32), timescale = 10000^(p/32)
// ---------------------------------------------------------------------------
__global__ __launch_bounds__(256) void rope_scale(float* __restrict__ Qd,
                                                  float* __restrict__ Kd,
                                                  const int* __restrict__ segpos) {
  const int PAIRS = BATCH * NUM_HEADS * SEQ * 32;  // per matrix
  int t = blockIdx.x * 256 + threadIdx.x;
  if (t >= 2 * PAIRS) return;
  float* ptr;
  float scale;
  if (t < PAIRS) { ptr = Qd; scale = 0.125f; }
  else           { ptr = Kd; scale = 1.0f; t -= PAIRS; }
  const int p  = t & 31;
  const int s  = (t >> 5) & (SEQ - 1);
  const int bh = t >> 16;  // SEQ*32 == 65536
  const int b  = bh >> 4;
  const float pos = (float)segpos[b * SEQ + s];
  // 10000^(-p/32) = exp2(-p * log2(10000)/32)
  const float inv = exp2f(-0.41524101186091903f * (float)p);
  const float ang = pos * inv;
  const float sn = sinf(ang);
  const float cs = cosf(ang);
  const size_t base = ((size_t)bh * SEQ + s) * HEAD_DIM;
  const float x1 = ptr[base + p];
  const float x2 = ptr[base + p + 32];
  ptr[base + p]      = (x1 * cs - x2 * sn) * scale;
  ptr[base + p + 32] = (x2 * cs + x1 * sn) * scale;
}

// ---------------------------------------------------------------------------
// Kernel 3: causal flash attention with softcap, per (b,h).
// Block = 128 threads (4 waves), 64-query tile, streams 64-key tiles.
// Writes X in [b, q, h*64+d] layout for the output projection GEMM.
// ---------------------------------------------------------------------------
__global__ __launch_bounds__(128) void attention(const float* __restrict__ Qd,
                                                 const float* __restrict__ Kd,
                                                 const float* __restrict__ Vd,
                                                 float* __restrict__ Xd) {
  __shared__ float Ks[64][65];      // [d][key]   (K transposed)
  __shared__ float Vs[64][65];      // [key][d]
  __shared__ float Ps[4][16][65];   // per-wave probability tile
  const int tid = threadIdx.x;
  const int lane = tid & 31;
  const int wave = tid >> 5;
  const int hv = lane >> 4;
  const int ln = lane & 15;
  const int q0 = blockIdx.x * 64;
  const int bh = blockIdx.y;
  const float* Qb = Qd + (size_t)bh * SEQ * HEAD_DIM;
  const float* Kb = Kd + (size_t)bh * SEQ * HEAD_DIM;
  const float* Vb = Vd + (size_t)bh * SEQ * HEAD_DIM;

  // This wave's 16 Q rows as A-fragments (held in registers for whole kernel)
  v2f qa[16];
  {
    const float* qrow = Qb + (size_t)(q0 + wave * 16 + ln) * HEAD_DIM + 2 * hv;
#pragma unroll
    for (int k4 = 0; k4 < 16; ++k4) qa[k4] = *(const v2f*)(qrow + k4 * 4);
  }

  v8f O[4];
#pragma unroll
  for (int n = 0; n < 4; ++n) O[n] = v8zero();
  float mrun[8], lrun[8];
#pragma unroll
  for (int r = 0; r < 8; ++r) { mrun[r] = -1e30f; lrun[r] = 0.0f; }

  const int jmax = blockIdx.x;  // last key tile index (causal)
  for (int j = 0; j <= jmax; ++j) {
    for (int i = tid; i < 64 * 64; i += 128) {
      const int r = i >> 6, c = i & 63;
      Ks[c][r] = Kb[(size_t)(j * 64 + r) * HEAD_DIM + c];
      Vs[r][c] = Vb[(size_t)(j * 64 + r) * HEAD_DIM + c];
    }
    if (j < jmax)
      __builtin_prefetch(Kb + (size_t)(j * 64 + 64 + (tid >> 1)) * HEAD_DIM, 0, 0);
    __syncthreads();

    // S(16x64) = Q(16x64) * K^T(64x64)
    v8f S[4];
#pragma unroll
    for (int n = 0; n < 4; ++n) S[n] = v8zero();
#pragma unroll
    for (int k4 = 0; k4 < 16; ++k4) {
      const int ka = k4 * 4 + 2 * hv;
#pragma unroll
      for (int n = 0; n < 4; ++n) {
        v2f bf;
        bf.x = Ks[ka][n * 16 + ln];
        bf.y = Ks[ka + 1][n * 16 + ln];
        S[n] = wmma_f32(qa[k4], bf, S[n]);
      }
    }

    // softcap + causal mask + online softmax (row = r + 8*hv within this wave)
#pragma unroll
    for (int r = 0; r < 8; ++r) {
      const int qrow = q0 + wave * 16 + r + 8 * hv;
      float mx = -1e30f;
#pragma unroll
      for (int n = 0; n < 4; ++n) {
        float sc = 50.0f * tanhf(S[n][r] * 0.02f);
        const int kcol = j * 64 + n * 16 + ln;
        if (kcol > qrow) sc = -50.0f;  // tanh(big_neg/50)*50
        S[n][r] = sc;
        mx = fmaxf(mx, sc);
      }
#pragma unroll
      for (int off = 8; off > 0; off >>= 1) mx = fmaxf(mx, __shfl_xor(mx, off));
      const float mnew = fmaxf(mrun[r], mx);
      const float corr = __expf(mrun[r] - mnew);
      float psum = 0.0f;
#pragma unroll
      for (int n = 0; n < 4; ++n) {
        const float pv = __expf(S[n][r] - mnew);
        S[n][r] = pv;
        psum += pv;
      }
#pragma unroll
      for (int off = 8; off > 0; off >>= 1) psum += __shfl_xor(psum, off);
      lrun[r] = lrun[r] * corr + psum;
      mrun[r] = mnew;
#pragma unroll
      for (int n = 0; n < 4; ++n) O[n][r] *= corr;
    }

    // P -> LDS (reshape C-layout into A-fragments)
#pragma unroll
    for (int n = 0; n < 4; ++n)
#pragma unroll
      for (int r = 0; r < 8; ++r)
        Ps[wave][r + 8 * hv][n * 16 + ln] = S[n][r];
    __syncthreads();

    // O(16x64) += P(16x64) * V(64x64)
#pragma unroll
    for (int k4 = 0; k4 < 16; ++k4) {
      const int ka = k4 * 4 + 2 * hv;
      v2f pa;
      pa.x = Ps[wave][ln][ka];
      pa.y = Ps[wave][ln][ka + 1];
#pragma unroll
      for (int n = 0; n < 4; ++n) {
        v2f bf;
        bf.x = Vs[ka][n * 16 + ln];
        bf.y = Vs[ka + 1][n * 16 + ln];
        O[n] = wmma_f32(pa, bf, O[n]);
      }
    }
    __syncthreads();
  }

  // normalize and store: X[b, q, h*64 + d]
  const int b = bh >> 4;
  const int h = bh & 15;
#pragma unroll
  for (int n = 0; n < 4; ++n)
#pragma unroll
    for (int r = 0; r < 8; ++r) {
      const int qrow = q0 + wave * 16 + r + 8 * hv;
      const int d = n * 16 + ln;
      Xd[(size_t)(b * SEQ + qrow) * DMODEL + h * HEAD_DIM + d] = O[n][r] / lrun[r];
    }
}

// ---------------------------------------------------------------------------
// Kernel 4: out projection  out[4096,1024] = X[4096,1024] * w_out[1024,1024]
// (w_out [H,64,M] flattens row-major to [1024,1024] with row = h*64+d)
// ---------------------------------------------------------------------------
__global__ __launch_bounds__(256) void gemm_out_proj(
    const float* __restrict__ A, const float* __restrict__ Bw,
    float* __restrict__ outp) {
  __shared__ float As[64][33];
  __shared__ float Bs[32][129];
  const int tid = threadIdx.x;
  const int lane = tid & 31;
  const int wave = tid >> 5;
  const int hv = lane >> 4;
  const int ln = lane & 15;
  const int m0 = blockIdx.y * 64;
  const int n0 = blockIdx.x * 128;
  const int mw = (wave >> 2) * 32;
  const int nw = (wave & 3) * 32;

  v8f C[2][2];
  C[0][0] = v8zero(); C[0][1] = v8zero();
  C[1][0] = v8zero(); C[1][1] = v8zero();

  for (int kb = 0; kb < DMODEL; kb += 32) {
    for (int i = tid; i < 512; i += 256) {
      const int r = i >> 3, c = (i & 7) << 2;
      const float4 v = *(const float4*)(A + (size_t)(m0 + r) * DMODEL + kb + c);
      As[r][c] = v.x; As[r][c + 1] = v.y; As[r][c + 2] = v.z; As[r][c + 3] = v.w;
    }
    for (int i = tid; i < 1024; i += 256) {
      const int r = i >> 5, c = (i & 31) << 2;
      const float4 v = *(const float4*)(Bw + (size_t)(kb + r) * DMODEL + n0 + c);
      Bs[r][c] = v.x; Bs[r][c + 1] = v.y; Bs[r][c + 2] = v.z; Bs[r][c + 3] = v.w;
    }
    __syncthreads();
#pragma unroll
    for (int kk = 0; kk < 32; kk += 4) {
      const int ka = kk + 2 * hv;
      v2f a0, a1, b0, b1;
      a0.x = As[mw + ln][ka];      a0.y = As[mw + ln][ka + 1];
      a1.x = As[mw + 16 + ln][ka]; a1.y = As[mw + 16 + ln][ka + 1];
      b0.x = Bs[ka][nw + ln];      b0.y = Bs[ka + 1][nw + ln];
      b1.x = Bs[ka][nw + 16 + ln]; b1.y = Bs[ka + 1][nw + 16 + ln];
      C[0][0] = wmma_f32(a0, b0, C[0][0]);
      C[0][1] = wmma_f32(a0, b1, C[0][1]);
      C[1][0] = wmma_f32(a1, b0, C[1][0]);
      C[1][1] = wmma_f32(a1, b1, C[1][1]);
    }
    __syncthreads();
  }

#pragma unroll
  for (int mi = 0; mi < 2; ++mi)
#pragma unroll
    for (int ni = 0; ni < 2; ++ni)
#pragma unroll
      for (int r = 0; r < 8; ++r) {
        const int m = m0 + mw + mi * 16 + r + 8 * hv;
        const int n = n0 + nw + ni * 16 + ln;
        outp[(size_t)m * DMODEL + n] = C[mi][ni][r];
      }
}

// ---------------------------------------------------------------------------
extern "C" void kernel_launch(void* const* d_in, const int* in_sizes, int n_in,
                              void* d_out, int out_size, void* d_ws, size_t ws_size,
                              hipStream_t stream) {
  const float* inputs = (const float*)d_in[0];   // [2,2048,1024] f32
  const int* segpos   = (const int*)d_in[1];     // [2,2048] i32
  // d_in[2] = mask (bool) — causal mask computed analytically, unused
  const float* w_in   = (const float*)d_in[3];   // [1024,16,192] f32
  const float* w_out  = (const float*)d_in[4];   // [16,64,1024] f32
  float* out = (float*)d_out;
  float* ws  = (float*)d_ws;

  const size_t NQKV = (size_t)BATCH * NUM_HEADS * SEQ * HEAD_DIM;  // 4,194,304
  float* Q = ws;
  float* K = ws + NQKV;
  float* V = ws + 2 * NQKV;
  float* X = ws + 3 * NQKV;  // total workspace: 64 MB

  gemm_in_proj<<<dim3(NPROJ / 128, (BATCH * SEQ) / 64), 256, 0, stream>>>(
      inputs, w_in, Q, K, V);
  const int total_pairs = 2 * BATCH * NUM_HEADS * SEQ * 32;
  rope_scale<<<(total_pairs + 255) / 256, 256, 0, stream>>>(Q, K, segpos);
  attention<<<dim3(SEQ / 64, BATCH * NUM_HEADS), 128, 0, stream>>>(Q, K, V, X);
  gemm_out_proj<<<dim3(DMODEL / 128, (BATCH * SEQ) / 64), 256, 0, stream>>>(
      X, w_out, out);
}